// VectorizedMINE_81930796139094
// MI455X (gfx1250) — compile-verified
//
#include <hip/hip_runtime.h>
#include <hip/hip_bf16.h>
#include <math.h>

typedef __attribute__((ext_vector_type(2))) float v2f;
typedef __attribute__((ext_vector_type(8))) float v8f;

// Problem constants (from reference): B=64, D=64, C=128, H=64
#define BB 64
#define DD 64
#define CC 128
#define HH 64
#define NPAIRS 8128          // C*(C-1)/2
#define ROWS 8192            // B*C

// ---------------------------------------------------------------------------
// Kernel 0: zero the two scalar accumulators (joint-sum, lse-sum)
// ---------------------------------------------------------------------------
__global__ void mine_init(float* __restrict__ acc) {
    if (threadIdx.x < 2) acc[threadIdx.x] = 0.0f;
}

// ---------------------------------------------------------------------------
// Kernel 1: WMMA f32 precompute.
//   P1[b*C+c, h] = sum_d x[b,d,c] * W1[d,     h]
//   P2[b*C+c, h] = sum_d x[b,d,c] * W1[64+d,  h]
// One wave computes one 16x16 tile of P{s} via 16 chained V_WMMA_F32_16X16X4_F32.
// Tiles: s in {0,1}, rowTile in [0,512), colTile in [0,4)  -> 4096 waves.
// Block = 128 threads (4 waves), grid = 1024.
// ---------------------------------------------------------------------------
__global__ __launch_bounds__(128) void mine_gemm_pre(const float* __restrict__ x,
                                                     const float* __restrict__ W1,
                                                     float* __restrict__ P) {
    const int tid  = threadIdx.x;
    const int lane = tid & 31;
    const int w    = blockIdx.x * 4 + (tid >> 5);  // global wave id, 0..4095
    const int s    = w >> 11;                      // which half of W1
    const int rem  = w & 2047;
    const int rt   = rem >> 2;                     // row tile, 0..511
    const int ct   = rem & 3;                      // col tile, 0..3
    const int half = lane >> 4;                    // 0: lanes 0-15, 1: lanes 16-31
    const int lm   = lane & 15;

    // A-fragment row (M) for this lane, mapped back to x[b, d, c]:
    const int r   = rt * 16 + lm;                  // global output row = b*C + c
    const int bb_ = r >> 7;                        // batch
    const int cc_ = r & 127;                       // channel
    const int ncol = ct * 16 + lm;                 // output column (h) for B-frag / store

    v8f acc = {};
    #pragma unroll
    for (int k0 = 0; k0 < DD; k0 += 4) {
        // A 16x4 f32 frag: VGPR v, half hl -> K = k0 + 2*hl + v ; row M = lm
        const int ka = k0 + half * 2;
        v2f a, bf;
        a.x = x[(bb_ * DD + ka    ) * CC + cc_];
        a.y = x[(bb_ * DD + ka + 1) * CC + cc_];
        // B 4x16 f32 frag: VGPR v, half hl -> K = k0 + 2*hl + v ; col N = lm
        const int kb = s * DD + ka;
        bf.x = W1[(kb    ) * HH + ncol];
        bf.y = W1[(kb + 1) * HH + ncol];
        acc = __builtin_amdgcn_wmma_f32_16x16x4_f32(false, a, false, bf,
                                                    (short)0, acc, false, false);
    }

    // D 16x16 f32: VGPR v -> row M = v + 8*half, col N = lm (within tile)
    float* out = P + (size_t)s * ROWS * HH;
    #pragma unroll
    for (int v = 0; v < 8; ++v) {
        const int gr = rt * 16 + v + half * 8;
        out[gr * HH + ncol] = acc[v];
    }
}

// ---------------------------------------------------------------------------
// Kernel 2: pair stage. One workgroup per batch b. Stages P1[b], P2[b],
// P2[perm[b]] in LDS (3 x 32KB). One wave per pair (i,j):
//   joint = relu(P1[b,i]+P2[b,j]+b1) . W2 + b2
//   marg  = relu(P1[b,i]+P2[perm[b],j]+b1) . W2 + b2   -> marg_ws[p*64 + b]
// Accumulates sum_p joint into acc[0] via one f32 atomic per wave.
// ---------------------------------------------------------------------------
__global__ __launch_bounds__(256) void mine_pair(const float* __restrict__ P,
                                                 const float* __restrict__ b1,
                                                 const float* __restrict__ W2,
                                                 const float* __restrict__ b2,
                                                 const int*   __restrict__ perm,
                                                 float* __restrict__ marg,
                                                 float* __restrict__ acc) {
    extern __shared__ float smem[];
    float* sA1  = smem;              // C*H = 8192 floats
    float* sA2  = smem + 8192;       // 8192
    float* sA2m = smem + 16384;      // 8192
    float* sW2  = smem + 24576;      // 64
    float* sB1  = sW2 + HH;          // 64

    const int b   = blockIdx.x;
    const int pb  = perm[b];
    const int tid = threadIdx.x;

    const float* A1g  = P + (size_t)b  * CC * HH;
    const float* A2g  = P + (size_t)ROWS * HH + (size_t)b  * CC * HH;
    const float* A2mg = P + (size_t)ROWS * HH + (size_t)pb * CC * HH;

    for (int i = tid; i < (CC * HH) / 4; i += 256) {
        ((float4*)sA1)[i]  = ((const float4*)A1g)[i];
        ((float4*)sA2)[i]  = ((const float4*)A2g)[i];
        ((float4*)sA2m)[i] = ((const float4*)A2mg)[i];
    }
    if (tid < HH) { sW2[tid] = W2[tid]; sB1[tid] = b1[tid]; }
    __syncthreads();

    const int lane = tid & 31;
    const int wid  = tid >> 5;
    const float w2x = sW2[lane * 2];
    const float w2y = sW2[lane * 2 + 1];
    const float b1x = sB1[lane * 2];
    const float b1y = sB1[lane * 2 + 1];
    const float bias2 = b2[0];

    float jointSum = 0.0f;
    for (int i = wid; i < CC - 1; i += 8) {
        const int pbase = i * (CC - 1) - (i * (i - 1)) / 2;  // first pair idx for row i
        const float2 a1 = *(const float2*)&sA1[i * HH + lane * 2];
        for (int j = i + 1; j < CC; ++j) {
            const float2 a2  = *(const float2*)&sA2[j * HH + lane * 2];
            const float2 a2m = *(const float2*)&sA2m[j * HH + lane * 2];
            float zx = fmaxf(a1.x + a2.x + b1x, 0.0f);
            float zy = fmaxf(a1.y + a2.y + b1y, 0.0f);
            float jp = zx * w2x + zy * w2y;
            float mx = fmaxf(a1.x + a2m.x + b1x, 0.0f);
            float my = fmaxf(a1.y + a2m.y + b1y, 0.0f);
            float mp = mx * w2x + my * w2y;
            #pragma unroll
            for (int off = 16; off > 0; off >>= 1) {
                jp += __shfl_xor(jp, off, 32);
                mp += __shfl_xor(mp, off, 32);
            }
            if (lane == 0) {
                jointSum += jp + bias2;
                const int p = pbase + (j - i - 1);
                marg[p * BB + b] = mp + bias2;
            }
        }
    }
    if (lane == 0) atomicAdd(&acc[0], jointSum);
}

// ---------------------------------------------------------------------------
// Kernel 3: per-pair logsumexp over the batch axis. One wave per pair,
// 2 batch values per lane, shuffle max + shuffle sum, atomic into acc[1].
// Grid = 8128 waves -> 1016 blocks of 256 threads.
// ---------------------------------------------------------------------------
__global__ __launch_bounds__(256) void mine_lse(const float* __restrict__ marg,
                                                float* __restrict__ acc) {
    const int lane = threadIdx.x & 31;
    const int p = (int)((blockIdx.x * blockDim.x + threadIdx.x) >> 5);
    if (p >= NPAIRS) return;
    const float v0 = marg[p * BB + lane];
    const float v1 = marg[p * BB + lane + 32];
    float mx = fmaxf(v0, v1);
    #pragma unroll
    for (int off = 16; off > 0; off >>= 1) mx = fmaxf(mx, __shfl_xor(mx, off, 32));
    float s = __expf(v0 - mx) + __expf(v1 - mx);
    #pragma unroll
    for (int off = 16; off > 0; off >>= 1) s += __shfl_xor(s, off, 32);
    if (lane == 0) atomicAdd(&acc[1], mx + __logf(s));
}

// ---------------------------------------------------------------------------
// Kernel 4: final scalar.
//  result = -SumJoint/(B*P) + SumLse/P - ln(B)
// ---------------------------------------------------------------------------
__global__ void mine_final(const float* __restrict__ acc, float* __restrict__ out) {
    if (threadIdx.x == 0 && blockIdx.x == 0) {
        const float Bf = (float)BB;
        const float Pf = (float)NPAIRS;
        out[0] = -acc[0] / (Bf * Pf) + acc[1] / Pf - logf(Bf);
    }
}

// ---------------------------------------------------------------------------
// Host launcher. Workspace layout (floats):
//   [0, 1048576)            : P1 | P2   (2 * 8192 * 64)
//   [1048576, 1568768)      : marginal matrix [8128][64]
//   [1568768, 1568770)      : accumulators {sumJoint, sumLse}
// Total ~6.27 MB.
// ---------------------------------------------------------------------------
extern "C" void kernel_launch(void* const* d_in, const int* in_sizes, int n_in,
                              void* d_out, int out_size, void* d_ws, size_t ws_size,
                              hipStream_t stream) {
    const float* x    = (const float*)d_in[0];
    const float* W1   = (const float*)d_in[1];
    const float* b1   = (const float*)d_in[2];
    const float* W2   = (const float*)d_in[3];
    const float* b2   = (const float*)d_in[4];
    const int*   perm = (const int*)  d_in[5];

    float* ws   = (float*)d_ws;
    float* P    = ws;                       // 2 * ROWS * HH
    float* marg = ws + 2 * ROWS * HH;       // NPAIRS * BB
    float* acc  = marg + NPAIRS * BB;       // 2 floats
    float* out  = (float*)d_out;

    mine_init<<<1, 32, 0, stream>>>(acc);

    // 4096 tile-waves, 4 waves/block -> 1024 blocks
    mine_gemm_pre<<<1024, 128, 0, stream>>>(x, W1, P);

    // one workgroup per batch; dynamic LDS = (3*8192 + 128) floats
    const size_t lds_bytes = (size_t)(3 * CC * HH + 2 * HH) * sizeof(float);
    mine_pair<<<BB, 256, lds_bytes, stream>>>(P, b1, W2, b2, perm, marg, acc);

    // 8128 pair-waves, 8 waves/block -> 1016 blocks
    mine_lse<<<NPAIRS / 8, 256, 0, stream>>>(marg, acc);

    mine_final<<<1, 1, 0, stream>>>(acc, out);
}